// caculatenet_55791625175129
// MI455X (gfx1250) — compile-verified
//
#include <hip/hip_runtime.h>

typedef __attribute__((ext_vector_type(2))) float v2f;
typedef __attribute__((ext_vector_type(8))) float v8f;

#define TLEN 600000
#define HID  64
#define ESTR 68  // padded LDS row stride (floats) for 64-wide matrices

struct Params {
  const float *traj1, *traj2;
  const float *ew, *eb, *qw, *qb, *kw, *kb;
  const float *fw1, *fb1, *fw2, *fb2, *fw3, *fb3;
  const float *ow1, *ob1, *ow2, *ob2, *ow3, *ob3, *ow4, *ob4;
  float* out;
  int out_base;
};

__device__ __forceinline__ v8f wmma4(v2f a, v2f b, v8f c) {
  // V_WMMA_F32_16X16X4_F32: full-precision fp32 matrix op on the WMMA pipe.
  return __builtin_amdgcn_wmma_f32_16x16x4_f32(false, a, false, b, (short)0, c,
                                               false, false);
}

// dst[16-row tile of wid] = src(tile) @ wgt(64x64) + bia, K=64 as 16 chained
// K=4 WMMA steps. A-operand re-read from LDS in the 16x4 fp32 A layout:
// lanes 0-15 hold K=0,1 / lanes 16-31 hold K=2,3 of row (lane&15).
__device__ __forceinline__ void project_qk(float* __restrict__ dst,
                                           const float* __restrict__ src,
                                           const float* __restrict__ wgt,
                                           const float* __restrict__ bia,
                                           int wid, int l16, int hi) {
#pragma unroll
  for (int n = 0; n < 4; ++n) {
    const int col = n * 16 + l16;
    v8f c = {};
#pragma unroll 4
    for (int kg = 0; kg < 16; ++kg) {
      const int kb = kg * 4 + hi * 2;
      const v2f a = *(const v2f*)(src + (wid * 16 + l16) * ESTR + kb);
      v2f b;
      b.x = wgt[(kb + 0) * HID + col];
      b.y = wgt[(kb + 1) * HID + col];
      c = wmma4(a, b, c);
    }
    const float bias = bia[col];
#pragma unroll
    for (int r = 0; r < 8; ++r)  // C/D layout: VGPR r -> row r (+8 for hi lanes)
      dst[(wid * 16 + r + hi * 8) * ESTR + col] = c[r] + bias;
  }
}

template <int W>
__global__ __launch_bounds__(128) void fused_window_kernel(Params p) {
  constexpr int G  = 60 / W;   // windows per block (exact for 5/10/15/20)
  constexpr int R  = 60;       // real rows per block (nw*W splits evenly)
  constexpr int FS = 2 * W;    // attention-feature stride
  constexpr int N1 = 4 * W;
  constexpr int N2 = 2 * W;
  constexpr int K1 = W * 32;

  extern __shared__ float smem[];
  float* Ea = smem;             // [64][ESTR] emb(traj1)
  float* Eb = Ea + 64 * ESTR;   // [64][ESTR] emb(traj2)
  float* Pq = Eb + 64 * ESTR;   // [64][ESTR] Q buffer (reused)
  float* Sk = Pq + 64 * ESTR;   // [64][ESTR] K buffer (reused)
  float* F  = Sk + 64 * ESTR;   // [64][FS]  attention features
  float* H1 = F + 64 * FS;      // [64][N1]
  float* H2 = H1 + 64 * N1;     // [64][N2]
  float* H3 = H2 + 64 * N2;     // [64][32]
  float* O1 = H3 + 64 * 32;     // [G][128]
  float* O2 = O1 + G * 128;     // [G][64]
  float* O3 = O2 + G * 64;      // [G][32]

  const int tid  = threadIdx.x;
  const int lane = tid & 31;
  const int wid  = tid >> 5;   // wave = 16-row WMMA tile
  const int l16  = lane & 15;
  const int hi   = lane >> 4;
  const int row0 = (int)blockIdx.x * R;

  // Warm L2 for the weight streams (lowers to global_prefetch_b8).
  if (wid == 0) {
    __builtin_prefetch(p.ow1 + (size_t)lane * 32, 0, 1);
    __builtin_prefetch(p.fw1 + (size_t)lane * 16, 0, 1);
  }

  // ---- Phase 1: embeddings = leaky_relu(traj @ ew + eb), K=4 == D_IN ----
  {
    int grow = row0 + wid * 16 + l16;
    grow = grow < TLEN ? grow : TLEN - 1;  // clamp pad rows (results unused)
    const v2f a1 = *(const v2f*)(p.traj1 + grow * 4 + hi * 2);
    const v2f a2 = *(const v2f*)(p.traj2 + grow * 4 + hi * 2);
#pragma unroll
    for (int n = 0; n < 4; ++n) {
      const int col = n * 16 + l16;
      v2f b;
      b.x = p.ew[(hi * 2 + 0) * HID + col];
      b.y = p.ew[(hi * 2 + 1) * HID + col];
      const float bias = p.eb[col];
      v8f c = {};
      c = wmma4(a1, b, c);
      v8f d = {};
      d = wmma4(a2, b, d);
#pragma unroll
      for (int r = 0; r < 8; ++r) {
        float x = c[r] + bias;
        Ea[(wid * 16 + r + hi * 8) * ESTR + col] = x > 0.f ? x : 0.01f * x;
        float y = d[r] + bias;
        Eb[(wid * 16 + r + hi * 8) * ESTR + col] = y > 0.f ? y : 0.01f * y;
      }
    }
  }
  __syncthreads();

  // ---- Pair 1: Qa = Ea@qw+qb, Kb = Eb@kw+kb ----
  project_qk(Pq, Ea, p.qw, p.qb, wid, l16, hi);
  project_qk(Sk, Eb, p.kw, p.kb, wid, l16, hi);
  __syncthreads();

  // attn_a logits: dot(Qa[i], Kb[j]) / sqrt(64) within each window.
  for (int idx = tid; idx < G * W * W; idx += 128) {
    const int w = idx / (W * W), ij = idx % (W * W);
    const int i = ij / W, j = ij % W;
    const float* q = Pq + (w * W + i) * ESTR;
    const float* k = Sk + (w * W + j) * ESTR;
    float acc = 0.f;
#pragma unroll 8
    for (int t = 0; t < HID; ++t) acc += q[t] * k[t];
    F[(w * W + i) * FS + j] = acc * 0.125f;
  }
  __syncthreads();

  // ---- Pair 2: Qb = Eb@qw+qb, Ka = Ea@kw+kb (reuse Pq/Sk) ----
  project_qk(Pq, Eb, p.qw, p.qb, wid, l16, hi);
  project_qk(Sk, Ea, p.kw, p.kb, wid, l16, hi);
  __syncthreads();

  for (int idx = tid; idx < G * W * W; idx += 128) {
    const int w = idx / (W * W), ij = idx % (W * W);
    const int i = ij / W, j = ij % W;
    const float* q = Pq + (w * W + i) * ESTR;
    const float* k = Sk + (w * W + j) * ESTR;
    float acc = 0.f;
#pragma unroll 8
    for (int t = 0; t < HID; ++t) acc += q[t] * k[t];
    F[(w * W + i) * FS + W + j] = acc * 0.125f;
  }
  __syncthreads();

  // Softmax per (row, half): each half normalized over its W columns.
  for (int t2 = tid; t2 < 2 * R; t2 += 128) {
    float* v = F + (t2 >> 1) * FS + (t2 & 1) * W;
    float m = v[0];
#pragma unroll
    for (int j = 1; j < W; ++j) m = fmaxf(m, v[j]);
    float s = 0.f;
#pragma unroll
    for (int j = 0; j < W; ++j) { const float e = __expf(v[j] - m); v[j] = e; s += e; }
    const float inv = 1.f / s;
#pragma unroll
    for (int j = 0; j < W; ++j) v[j] *= inv;
  }
  __syncthreads();

  // ---- fc stack: [R,2W] -> relu -> [R,4W] -> relu -> [R,2W] -> relu -> [R,32]
  for (int idx = tid; idx < R * N1; idx += 128) {
    const int r = idx / N1, c = idx % N1;
    float acc = p.fb1[c];
    for (int k = 0; k < FS; ++k) acc += F[r * FS + k] * p.fw1[k * N1 + c];
    H1[idx] = acc > 0.f ? acc : 0.f;
  }
  __syncthreads();
  for (int idx = tid; idx < R * N2; idx += 128) {
    const int r = idx / N2, c = idx % N2;
    float acc = p.fb2[c];
    for (int k = 0; k < N1; ++k) acc += H1[r * N1 + k] * p.fw2[k * N2 + c];
    H2[idx] = acc > 0.f ? acc : 0.f;
  }
  __syncthreads();
  for (int idx = tid; idx < R * 32; idx += 128) {
    const int r = idx >> 5, c = idx & 31;
    float acc = p.fb3[c];
    for (int k = 0; k < N2; ++k) acc += H2[r * N2 + k] * p.fw3[k * 32 + c];
    H3[idx] = acc > 0.f ? acc : 0.f;
  }
  __syncthreads();

  // ---- out MLP per window: [W*32] -> 128 -> 64 -> 32 -> 1 ----
  for (int idx = tid; idx < G * 128; idx += 128) {
    const int wv = idx >> 7, c = idx & 127;
    float acc = p.ob1[c];
    const float* fin = H3 + wv * W * 32;  // contiguous reshape of window rows
    for (int k = 0; k < K1; ++k) acc += fin[k] * p.ow1[k * 128 + c];
    O1[idx] = acc > 0.f ? acc : 0.f;
  }
  __syncthreads();
  for (int idx = tid; idx < G * 64; idx += 128) {
    const int wv = idx >> 6, c = idx & 63;
    float acc = p.ob2[c];
    for (int k = 0; k < 128; ++k) acc += O1[wv * 128 + k] * p.ow2[k * 64 + c];
    O2[idx] = acc > 0.f ? acc : 0.f;
  }
  __syncthreads();
  for (int idx = tid; idx < G * 32; idx += 128) {
    const int wv = idx >> 5, c = idx & 31;
    float acc = p.ob3[c];
    for (int k = 0; k < 64; ++k) acc += O2[wv * 64 + k] * p.ow3[k * 32 + c];
    O3[idx] = acc > 0.f ? acc : 0.f;
  }
  __syncthreads();
  for (int wv = tid; wv < G; wv += 128) {
    float acc = p.ob4[0];
    for (int k = 0; k < 32; ++k) acc += O3[wv * 32 + k] * p.ow4[k];
    p.out[p.out_base + (int)blockIdx.x * G + wv] = acc;
  }
}

// ---------------- host side ----------------
namespace {

struct Map { int t1, t2, emb, wq, wk, fc[4], out[4]; };

bool checkFc(const int* s, int b, int W) {
  return s[b] == 2 * W * 4 * W && s[b + 1] == 4 * W && s[b + 2] == 4 * W * 2 * W &&
         s[b + 3] == 2 * W && s[b + 4] == 2 * W * 32 && s[b + 5] == 32;
}
bool checkOut(const int* s, int b, int W) {
  return s[b] == W * 32 * 128 && s[b + 1] == 128 && s[b + 2] == 8192 &&
         s[b + 3] == 64 && s[b + 4] == 2048 && s[b + 5] == 32 &&
         s[b + 6] == 32 && s[b + 7] == 1;
}
bool verifyMap(const Map& m, const int* s, int n) {
  static const int Ws[4] = {5, 10, 15, 20};
  if (n < 64) return false;
  if (s[m.t1] != TLEN * 4 || s[m.t2] != TLEN * 4) return false;
  if (s[m.emb] != 256 || s[m.emb + 1] != 64) return false;
  if (s[m.wq] != 4096 || s[m.wq + 1] != 64) return false;
  if (s[m.wk] != 4096 || s[m.wk + 1] != 64) return false;
  for (int i = 0; i < 4; ++i)
    if (!checkFc(s, m.fc[i], Ws[i]) || !checkOut(s, m.out[i], Ws[i])) return false;
  return true;
}

}  // namespace

extern "C" void kernel_launch(void* const* d_in, const int* in_sizes, int n_in,
                              void* d_out, int out_size, void* d_ws, size_t ws_size,
                              hipStream_t stream) {
  (void)out_size; (void)d_ws; (void)ws_size;

  // Candidate flattenings of setup_inputs(): insertion order, trajs-first with
  // sorted params, and fully-sorted pytree order. Pick by size signature.
  static const Map candA = {0, 1, 2, 4, 6, {8, 22, 36, 50}, {14, 28, 42, 56}};
  static const Map candB = {0, 1, 2, 62, 60, {22, 4, 10, 16}, {52, 28, 36, 44}};
  static const Map candC = {62, 63, 0, 60, 58, {20, 2, 8, 14}, {50, 26, 34, 42}};
  const Map* cands[3] = {&candA, &candB, &candC};
  const Map* m = &candA;
  for (int i = 0; i < 3; ++i)
    if (verifyMap(*cands[i], in_sizes, n_in)) { m = cands[i]; break; }

  auto in = [&](int i) { return (const float*)d_in[i]; };
  static const int Ws[4]   = {5, 10, 15, 20};
  static const int base[4] = {0, 120000, 180000, 220000};

  for (int i = 0; i < 4; ++i) {
    Params pp;
    pp.traj1 = in(m->t1); pp.traj2 = in(m->t2);
    pp.ew = in(m->emb); pp.eb = in(m->emb + 1);
    pp.qw = in(m->wq);  pp.qb = in(m->wq + 1);
    pp.kw = in(m->wk);  pp.kb = in(m->wk + 1);
    const int fb = m->fc[i];
    pp.fw1 = in(fb);     pp.fb1 = in(fb + 1);
    pp.fw2 = in(fb + 2); pp.fb2 = in(fb + 3);
    pp.fw3 = in(fb + 4); pp.fb3 = in(fb + 5);
    const int ob = m->out[i];
    pp.ow1 = in(ob);     pp.ob1 = in(ob + 1);
    pp.ow2 = in(ob + 2); pp.ob2 = in(ob + 3);
    pp.ow3 = in(ob + 4); pp.ob3 = in(ob + 5);
    pp.ow4 = in(ob + 6); pp.ob4 = in(ob + 7);
    pp.out = (float*)d_out;
    pp.out_base = base[i];

    const int W = Ws[i];
    const size_t sm =
        (size_t)(4 * 64 * ESTR + 64 * (8 * W + 32) + (60 / W) * 224) * sizeof(float);
    switch (W) {
      case 5:  fused_window_kernel<5><<<10000, 128, sm, stream>>>(pp); break;
      case 10: fused_window_kernel<10><<<10000, 128, sm, stream>>>(pp); break;
      case 15: fused_window_kernel<15><<<10000, 128, sm, stream>>>(pp); break;
      case 20: fused_window_kernel<20><<<10000, 128, sm, stream>>>(pp); break;
    }
  }
}